// HistoryAttentionNet_26886495272963
// MI455X (gfx1250) — compile-verified
//
#include <hip/hip_runtime.h>
#include <hip/hip_bf16.h>
#include <math.h>

#define HIDDEN    768
#define SEQ_LEN   512
#define ROWS      64
#define SLICES    16
#define MAX_TURNS 11

typedef __attribute__((ext_vector_type(2))) float v2f;
typedef __attribute__((ext_vector_type(8))) float v8f;

// ---------------------------------------------------------------------------
// Kernel 1 (tiny): logits -> per-segment softmax -> w[r], seg[r], probs[s,t]
// ---------------------------------------------------------------------------
__global__ __launch_bounds__(256) void probs_kernel(
    const float* __restrict__ hist,        // (64, 768)
    const float* __restrict__ W,           // (1, 768)
    const float* __restrict__ b,           // (1,)
    const int*   __restrict__ slice_mask,  // (64,) use first 16
    float* __restrict__ w_out,             // (64,) workspace
    int*   __restrict__ seg_out,           // (64,) workspace
    float* __restrict__ probs_out)         // (16*11,)
{
  __shared__ float l[ROWS];
  __shared__ float sums[SLICES];
  __shared__ int   off_s[SLICES], ns_s[SLICES], pad_s[SLICES];

  const int tid  = threadIdx.x;
  const int lane = tid & 31;
  const int wv   = tid >> 5;            // 8 waves

  // 64 dot products of length 768: 8 rows per wave, wave32 shuffle reduction.
  for (int i = 0; i < 8; ++i) {
    int r = wv * 8 + i;
    float p = 0.f;
    for (int h = lane; h < HIDDEN; h += 32)
      p += hist[r * HIDDEN + h] * W[h];
    for (int o = 16; o > 0; o >>= 1)
      p += __shfl_xor(p, o, 32);
    if (lane == 0) l[r] = p + b[0];
  }
  __syncthreads();

  // Per-slice offsets / pad lengths / softmax denominators.
  if (tid < SLICES) {
    int off = 0;
    for (int s = 0; s < tid; ++s) off += slice_mask[s];
    int ns = slice_mask[tid];
    off_s[tid] = off;
    ns_s[tid]  = ns;
    pad_s[tid] = MAX_TURNS - ns;
    float se = 0.f;
    for (int j = 0; j < ns; ++j) {
      int r = off + j;
      if (r >= 0 && r < ROWS) se += expf(l[r]);
    }
    sums[tid] = se;
  }
  __syncthreads();

  // Per-row weight and segment id.
  if (tid < ROWS) {
    int sg = 0;
    for (int s = 0; s < SLICES; ++s)
      if (tid >= off_s[s] && tid < off_s[s] + ns_s[s]) sg = s;
    seg_out[tid] = sg;
    w_out[tid]   = expf(l[tid]) / sums[sg];
  }

  // probs (16 x 11), zero where invalid.
  if (tid < SLICES * MAX_TURNS) {
    int s = tid / MAX_TURNS, t = tid % MAX_TURNS;
    float pv = 0.f;
    int pad = pad_s[s];
    if (t >= pad) {
      int r = off_s[s] + t - pad;
      if (r >= 0 && r < ROWS) pv = expf(l[r]) / sums[s];
    }
    probs_out[tid] = pv;
  }
}

// ---------------------------------------------------------------------------
// Kernel 2: segment-sum as D(16xN) = M(16x64) @ X(64xN) via WMMA f32 16x16x4.
// N is a compile-time row stride so B-fragment loads fold to one base pointer
// plus immediate offsets (N*4 = 1.5 MB fits the signed 24-bit IOFFSET), and
// the K-loop advances the pointer with a single uniform 64-bit add per step.
// A-frag (16x4 f32): v0 = K=0 (lanes 0-15, M=lane) & K=2 (lanes 16-31);
// v1 = K=1 & K=3. B-frag (4x16) mirrors the K split. C/D: VGPR v holds
// rows v / v+8 across lane halves.
// ---------------------------------------------------------------------------
template <int N>
__global__ __launch_bounds__(256) void seg_wmma_kernel(
    const float* __restrict__ B,      // (64, N)
    const float* __restrict__ w,      // (64,)
    const int*   __restrict__ seg,    // (64,)
    float* __restrict__ D)            // (16, N)
{
  const int nTiles = N / 16;
  int wave = (int)((blockIdx.x * blockDim.x + threadIdx.x) >> 5);
  const int lane = threadIdx.x & 31;
  if (wave >= nTiles) return;          // uniform per-wave branch: EXEC stays all-1s

  const int kk  = (lane < 16) ? 0 : 2; // K sub-index for this lane half
  const int m   = lane & 15;           // M (A) / N (B,D) index within tile
  const int col = wave * 16 + m;

  // Build all 16 A-fragments from (w, seg): M[s, r] = (seg[r]==s) ? w[r] : 0.
  v2f afrag[16];
#pragma unroll
  for (int c = 0; c < 16; ++c) {
    int r0 = 4 * c + kk;
    afrag[c].x = (seg[r0]     == m) ? w[r0]     : 0.0f;
    afrag[c].y = (seg[r0 + 1] == m) ? w[r0 + 1] : 0.0f;
  }

  const float* p = B + (size_t)kk * N + col;   // lane base pointer
  v8f acc = {0.f, 0.f, 0.f, 0.f, 0.f, 0.f, 0.f, 0.f};
#pragma unroll
  for (int c = 0; c < 16; ++c) {
    v2f bfrag;
    bfrag.x = p[0];            // row 4c+kk     -> immediate offset 0
    bfrag.y = p[(size_t)N];    // row 4c+kk+1   -> immediate offset N*4
    // 8 args: (neg_a, A, neg_b, B, c_mod, C, reuse_a, reuse_b)
    acc = __builtin_amdgcn_wmma_f32_16x16x4_f32(
        false, afrag[c], false, bfrag, (short)0, acc, false, false);
    p += (size_t)4 * N;        // one uniform 64-bit add per K-step
  }

  // Store D tile: VGPR v holds row v (lanes 0-15) and row v+8 (lanes 16-31).
  float* q = D + (size_t)((lane < 16) ? 0 : 8) * N + col;
#pragma unroll
  for (int v = 0; v < 8; ++v)
    q[(size_t)v * N] = acc[v];
}

// ---------------------------------------------------------------------------
extern "C" void kernel_launch(void* const* d_in, const int* in_sizes, int n_in,
                              void* d_out, int out_size, void* d_ws, size_t ws_size,
                              hipStream_t stream) {
  const float* bert       = (const float*)d_in[0];  // (64,512,768)
  const float* hist       = (const float*)d_in[1];  // (64,768)
  const float* mtl        = (const float*)d_in[2];  // (64,768)
  const float* W          = (const float*)d_in[3];  // (1,768)
  const float* b          = (const float*)d_in[4];  // (1,)
  const int*   slice_mask = (const int*)d_in[5];    // (64,)
  // d_in[6] = slice_num (==16, fixed at compile time)

  float* bert_out  = (float*)d_out;                                   // 16*512*768
  float* mtl_out   = bert_out + (size_t)SLICES * SEQ_LEN * HIDDEN;    // 16*768
  float* probs_out = mtl_out + (size_t)SLICES * HIDDEN;               // 16*11

  float* w_ws   = (float*)d_ws;
  int*   seg_ws = (int*)((char*)d_ws + ROWS * sizeof(float));

  probs_kernel<<<1, 256, 0, stream>>>(hist, W, b, slice_mask,
                                      w_ws, seg_ws, probs_out);

  constexpr int NB = SEQ_LEN * HIDDEN;          // 393216 -> 24576 tiles
  constexpr int NM = HIDDEN;                    // 768    -> 48 tiles
  constexpr int wavesPerBlock = 256 / 32;

  seg_wmma_kernel<NB><<<(NB / 16 + wavesPerBlock - 1) / wavesPerBlock, 256, 0, stream>>>(
      bert, w_ws, seg_ws, bert_out);
  seg_wmma_kernel<NM><<<(NM / 16 + wavesPerBlock - 1) / wavesPerBlock, 256, 0, stream>>>(
      mtl, w_ws, seg_ws, mtl_out);
}